// AttnDecoderRNN_44839458570800
// MI455X (gfx1250) — compile-verified
//
#include <hip/hip_runtime.h>
#include <math.h>

typedef float v2f __attribute__((ext_vector_type(2)));
typedef float v8f __attribute__((ext_vector_type(8)));

#define H 1024
#define V 50257
#define L 512

// workspace layout (float offsets, all 16B-aligned)
#define WS_XCAT0   0        // 2048: [embedded | h0]
#define WS_ASCORE  2048     // 512 : raw attention scores
#define WS_ATTNW   2560     // 512 : softmax weights
#define WS_XCAT1   3072     // 2048: [embedded | attn_applied]
#define WS_XVEC    5120     // 1024: relu(comb)
#define WS_GI      6144     // 3072
#define WS_GH      9216     // 3072
#define WS_HNEW    12288    // 1024
#define WS_RED     13312    // 16  : [max, log(sumexp)]
#define WS_LOGITS  13328    // 50257

// ---------------- kernel 0: xcat0 = [emb[id] | h0] ----------------
__global__ void k_prep(const int* __restrict__ ids, const float* __restrict__ hidden,
                       const float* __restrict__ emb, float* __restrict__ ws) {
  int k = blockIdx.x * blockDim.x + threadIdx.x;
  if (k < 2 * H) {
    int id = ids[0];
    ws[WS_XCAT0 + k] = (k < H) ? emb[(size_t)id * H + k] : hidden[k - H];
  }
}

// ------------- generic 16-rows-per-wave WMMA f32 mat-vec -------------
// y[r] = dot(W[r,0:K], x) + bias[r], optional ReLU. K % 4 == 0.
template <int ACT>
__global__ void k_matvec_wmma(const float* __restrict__ W, const float* __restrict__ x,
                              const float* __restrict__ bias, float* __restrict__ y,
                              int nrows, int K) {
  const int wave = blockIdx.x * (blockDim.x >> 5) + (threadIdx.x >> 5);
  const int row0 = wave * 16;
  if (row0 >= nrows) return;                 // whole-wave exit: EXEC stays all-1s for WMMA

  const int lane = threadIdx.x & 31;
  const int m    = lane & 15;
  const int koff = (lane >> 4) << 1;         // 0 for lanes 0-15, 2 for lanes 16-31

  int rowA = row0 + m;
  if (rowA >= nrows) rowA = nrows - 1;       // clamp (garbage rows never stored)
  const float* __restrict__ wrow = W + (size_t)rowA * K + koff;
  const float* __restrict__ xk   = x + koff;

  v8f acc = {0.f, 0.f, 0.f, 0.f, 0.f, 0.f, 0.f, 0.f};
#pragma unroll 4
  for (int k0 = 0; k0 < K; k0 += 4) {
    v2f a = *(const v2f*)(wrow + k0);        // A[m][k0+koff .. +1]
    v2f b = *(const v2f*)(xk + k0);          // B[k][n] = x chunk, replicated over n
    acc = __builtin_amdgcn_wmma_f32_16x16x4_f32(
        /*neg_a=*/false, a, /*neg_b=*/false, b,
        /*c_mod=*/(short)0, acc, /*reuse_a=*/false, /*reuse_b=*/false);
  }

  // column N=0 lives in lanes 0 (rows 0-7) and 16 (rows 8-15)
  if ((lane & 15) == 0) {
    const int base = row0 + ((lane >> 4) << 3);
#pragma unroll
    for (int i = 0; i < 8; ++i) {
      int r = base + i;
      if (r < nrows) {
        float v = acc[i] + bias[r];
        if (ACT) v = fmaxf(v, 0.f);
        y[r] = v;
      }
    }
  }
}

// ---------------- kernel: softmax over L=512 scores ----------------
__global__ void k_softmax(float* __restrict__ ws, float* __restrict__ out) {
  __shared__ float s[L];
  const int t = threadIdx.x;
  float v = ws[WS_ASCORE + t];
  s[t] = v; __syncthreads();
  for (int off = L / 2; off > 0; off >>= 1) {
    if (t < off) s[t] = fmaxf(s[t], s[t + off]);
    __syncthreads();
  }
  const float mx = s[0]; __syncthreads();
  const float e = __expf(v - mx);
  s[t] = e; __syncthreads();
  for (int off = L / 2; off > 0; off >>= 1) {
    if (t < off) s[t] += s[t + off];
    __syncthreads();
  }
  const float w = e / s[0];
  ws[WS_ATTNW + t] = w;
  out[V + H + t] = w;                         // attn_weights output
}

// ------- kernel: attn_applied = w @ enc; build xcat1 -------
__global__ void k_attn_apply(const float* __restrict__ enc, float* __restrict__ ws) {
  __shared__ float w[L];
  const int t = threadIdx.x;
  for (int i = t; i < L; i += blockDim.x) w[i] = ws[WS_ATTNW + i];
  __syncthreads();
  const int h = blockIdx.x * blockDim.x + t;
  if (h < H) {
    float acc = 0.f;
#pragma unroll 8
    for (int l = 0; l < L; ++l) acc = fmaf(w[l], enc[(size_t)l * H + h], acc);
    ws[WS_XCAT1 + h]     = ws[WS_XCAT0 + h];  // embedded
    ws[WS_XCAT1 + H + h] = acc;               // attn_applied
  }
}

// ---------------- kernel: GRU gate math ----------------
__global__ void k_gru(const float* __restrict__ hidden, float* __restrict__ ws,
                      float* __restrict__ out) {
  const int j = blockIdx.x * blockDim.x + threadIdx.x;
  if (j < H) {
    const float r = 1.f / (1.f + __expf(-(ws[WS_GI + j] + ws[WS_GH + j])));
    const float z = 1.f / (1.f + __expf(-(ws[WS_GI + H + j] + ws[WS_GH + H + j])));
    const float n = tanhf(ws[WS_GI + 2 * H + j] + r * ws[WS_GH + 2 * H + j]);
    const float hn = (1.f - z) * n + z * hidden[j];
    ws[WS_HNEW + j] = hn;
    out[V + j] = hn;                          // h_new output
  }
}

// -------- kernel: max & log-sum-exp over 50257 logits (one block) --------
__global__ void k_logsumexp(float* __restrict__ ws) {
  __shared__ float s[1024];
  const int t = threadIdx.x;
  float mx = -3.402823466e38f;
  for (int i = t; i < V; i += 1024) mx = fmaxf(mx, ws[WS_LOGITS + i]);
  s[t] = mx; __syncthreads();
  for (int off = 512; off > 0; off >>= 1) {
    if (t < off) s[t] = fmaxf(s[t], s[t + off]);
    __syncthreads();
  }
  mx = s[0]; __syncthreads();
  float sum = 0.f;
  for (int i = t; i < V; i += 1024) sum += __expf(ws[WS_LOGITS + i] - mx);
  s[t] = sum; __syncthreads();
  for (int off = 512; off > 0; off >>= 1) {
    if (t < off) s[t] += s[t + off];
    __syncthreads();
  }
  if (t == 0) { ws[WS_RED] = mx; ws[WS_RED + 1] = logf(s[0]); }
}

// ---------------- kernel: finalize log-softmax ----------------
__global__ void k_finalize(const float* __restrict__ ws, float* __restrict__ out) {
  const int i = blockIdx.x * blockDim.x + threadIdx.x;
  if (i < V) out[i] = ws[WS_LOGITS + i] - ws[WS_RED] - ws[WS_RED + 1];
}

extern "C" void kernel_launch(void* const* d_in, const int* in_sizes, int n_in,
                              void* d_out, int out_size, void* d_ws, size_t ws_size,
                              hipStream_t stream) {
  const int*   ids    = (const int*)d_in[0];
  const float* hidden = (const float*)d_in[1];
  const float* enc    = (const float*)d_in[2];
  const float* emb    = (const float*)d_in[3];
  const float* attn_W = (const float*)d_in[4];
  const float* attn_b = (const float*)d_in[5];
  const float* comb_W = (const float*)d_in[6];
  const float* comb_b = (const float*)d_in[7];
  const float* W_ih   = (const float*)d_in[8];
  const float* W_hh   = (const float*)d_in[9];
  const float* b_ih   = (const float*)d_in[10];
  const float* b_hh   = (const float*)d_in[11];
  const float* out_W  = (const float*)d_in[12];
  const float* out_b  = (const float*)d_in[13];
  float* out = (float*)d_out;
  float* ws  = (float*)d_ws;

  const int WPB = 4, TPB = WPB * 32;          // 4 waves per block
  auto grid_for = [&](int rows) { int tiles = (rows + 15) / 16; return (tiles + WPB - 1) / WPB; };

  // 0) xcat0 = [emb[id] | h0]
  k_prep<<<(2 * H + 255) / 256, 256, 0, stream>>>(ids, hidden, emb, ws);

  // 1) attention scores: (512 x 2048) . xcat0
  k_matvec_wmma<0><<<grid_for(L), TPB, 0, stream>>>(attn_W, ws + WS_XCAT0, attn_b,
                                                    ws + WS_ASCORE, L, 2 * H);
  // 2) softmax -> attn_weights (also to d_out)
  k_softmax<<<1, L, 0, stream>>>(ws, out);

  // 3) attn_applied = w @ enc ; build xcat1
  k_attn_apply<<<(H + 255) / 256, 256, 0, stream>>>(enc, ws);

  // 4) x = relu(comb_W . xcat1 + comb_b)   (1024 x 2048)
  k_matvec_wmma<1><<<grid_for(H), TPB, 0, stream>>>(comb_W, ws + WS_XCAT1, comb_b,
                                                    ws + WS_XVEC, H, 2 * H);
  // 5) gi = W_ih . x + b_ih ; gh = W_hh . h0 + b_hh   (3072 x 1024 each)
  k_matvec_wmma<0><<<grid_for(3 * H), TPB, 0, stream>>>(W_ih, ws + WS_XVEC, b_ih,
                                                        ws + WS_GI, 3 * H, H);
  k_matvec_wmma<0><<<grid_for(3 * H), TPB, 0, stream>>>(W_hh, hidden, b_hh,
                                                        ws + WS_GH, 3 * H, H);
  // 6) GRU combine -> h_new (also to d_out)
  k_gru<<<(H + 255) / 256, 256, 0, stream>>>(hidden, ws, out);

  // 7) logits = out_W . h_new + out_b   (50257 x 1024)  -- dominant, BW-bound
  k_matvec_wmma<0><<<grid_for(V), TPB, 0, stream>>>(out_W, ws + WS_HNEW, out_b,
                                                    ws + WS_LOGITS, V, H);
  // 8) log-softmax
  k_logsumexp<<<1, 1024, 0, stream>>>(ws);
  k_finalize<<<(V + 255) / 256, 256, 0, stream>>>(ws, out);
}